// GCGRU_44839458570786
// MI455X (gfx1250) — compile-verified
//
#include <hip/hip_runtime.h>
#include <math.h>

// ---------------- problem constants ----------------
#define CB    8
#define CG    2048
#define CN    16384        // B*G nodes
#define CFIN  32
#define CH    64
#define CT    10
#define CE    262144       // edges
#define K25   25
#define KTOT  26           // 25 spline kernels + 1 root block
#define LDW   (KTOT * CH)  // 1664, packed weight / xw row stride

typedef float v2f __attribute__((ext_vector_type(2)));
typedef float v8f __attribute__((ext_vector_type(8)));

// ---------------- small helpers ----------------
__global__ __launch_bounds__(256) void zero_f32(float* __restrict__ p, int n) {
    int i = blockIdx.x * 256 + threadIdx.x;
    if (i < n) p[i] = 0.0f;
}

// per-edge: src/dst, spline basis + weight indices, degree accumulation
__global__ __launch_bounds__(256)
void prep_edges(const int* __restrict__ eidx, const float* __restrict__ eattr,
                int* __restrict__ src, int* __restrict__ dst,
                float* __restrict__ basis, int* __restrict__ widx,
                float* __restrict__ deg) {
    int e = blockIdx.x * 256 + threadIdx.x;
    if (e >= CE) return;
    int s = eidx[e];
    int d = eidx[CE + e];
    src[e] = s;
    dst[e] = d;
    __hip_atomic_fetch_add(&deg[d], 1.0f, __ATOMIC_RELAXED, __HIP_MEMORY_SCOPE_AGENT);
    float f0 = eattr[e * 2 + 0] * 4.0f;   // (M-1) = 4
    float f1 = eattr[e * 2 + 1] * 4.0f;
    float fl0 = fminf(fmaxf(floorf(f0), 0.0f), 3.0f);
    float fl1 = fminf(fmaxf(floorf(f1), 0.0f), 3.0f);
    float t0 = f0 - fl0, t1 = f1 - fl1;
    int k0 = (int)fl0, k1 = (int)fl1;
#pragma unroll
    for (int sIdx = 0; sIdx < 4; ++sIdx) {
        int b0 = (sIdx >> 1) & 1;  // dim0 bit (itertools.product order)
        int b1 = sIdx & 1;         // dim1 bit
        float w = (b0 ? t0 : 1.0f - t0) * (b1 ? t1 : 1.0f - t1);
        basis[e * 4 + sIdx] = w;
        widx[e * 4 + sIdx] = (k0 + b0) * 1 + (k1 + b1) * 5;  // strides (1,5)
    }
}

__global__ __launch_bounds__(256) void inv_deg_k(float* __restrict__ deg) {
    int i = blockIdx.x * 256 + threadIdx.x;
    if (i < CN) deg[i] = 1.0f / fmaxf(deg[i], 1.0f);
}

// x[B,G,FIN,T] -> xs[T, N, FIN]
__global__ __launch_bounds__(256)
void transpose_x(const float* __restrict__ x, float* __restrict__ xs) {
    int i = blockIdx.x * 256 + threadIdx.x;  // < T*N*FIN
    int t = i / (CN * CFIN);
    int rem = i - t * (CN * CFIN);
    int n = rem >> 5;
    int f = rem & 31;
    xs[i] = x[((size_t)n * CFIN + f) * CT + t];
}

// W[K,F,H] + root[F,H]  ->  Wc[F, KTOT*H]  (root as k-block 25)
__global__ __launch_bounds__(256)
void repack_w(const float* __restrict__ Wk, const float* __restrict__ root,
              float* __restrict__ wc, int F) {
    int i = blockIdx.x * 256 + threadIdx.x;
    if (i >= F * LDW) return;
    int f = i / LDW;
    int c = i - f * LDW;
    int k = c >> 6;
    int o = c & 63;
    wc[i] = (k < K25) ? Wk[((size_t)k * F + f) * CH + o] : root[(size_t)f * CH + o];
}

// ---------------- WMMA fp32 GEMM: D[16384,1664] = X[16384,F] * W[F,1664] ----------------
// grid (128, 26), block 256 (8 waves). One wave = 16x64 output tile.
// A 16x4 f32 layout: VGPR0 = K{0,2}, VGPR1 = K{1,3} (lanes 0-15 / 16-31 = M rows).
// B 4x16 layout: row-striped across lanes; VGPRv holds rows {v, v+2}.
template <int F>
__global__ __launch_bounds__(256)
void gemm_wmma(const float* __restrict__ X, const float* __restrict__ W,
               float* __restrict__ D) {
    const int lane = threadIdx.x & 31;
    const int wave = threadIdx.x >> 5;
    const int l16  = lane & 15;
    const int half = lane >> 4;
    const int rowBase = (blockIdx.x * 8 + wave) * 16;
    const int colBase = blockIdx.y * 64;

    v8f c0 = {}, c1 = {}, c2 = {}, c3 = {};
    const float* xrow = X + (size_t)(rowBase + l16) * F;

    for (int kk = 0; kk < F; kk += 4) {
        v2f a;
        a.x = xrow[kk + 2 * half + 0];
        a.y = xrow[kk + 2 * half + 1];
        const float* w0 = W + (size_t)(kk + 2 * half + 0) * LDW + colBase + l16;
        const float* w1 = W + (size_t)(kk + 2 * half + 1) * LDW + colBase + l16;
        v2f b0, b1, b2, b3;
        b0.x = w0[0];  b0.y = w1[0];
        b1.x = w0[16]; b1.y = w1[16];
        b2.x = w0[32]; b2.y = w1[32];
        b3.x = w0[48]; b3.y = w1[48];
        c0 = __builtin_amdgcn_wmma_f32_16x16x4_f32(false, a, false, b0, (short)0, c0, false, false);
        c1 = __builtin_amdgcn_wmma_f32_16x16x4_f32(false, a, false, b1, (short)0, c1, false, false);
        c2 = __builtin_amdgcn_wmma_f32_16x16x4_f32(false, a, false, b2, (short)0, c2, false, false);
        c3 = __builtin_amdgcn_wmma_f32_16x16x4_f32(false, a, false, b3, (short)0, c3, false, false);
    }

    float* drow = D + (size_t)rowBase * LDW + colBase;
#pragma unroll
    for (int r = 0; r < 8; ++r) {
        int m = r + 8 * half;
        drow[(size_t)m * LDW +  0 + l16] = c0[r];
        drow[(size_t)m * LDW + 16 + l16] = c1[r];
        drow[(size_t)m * LDW + 32 + l16] = c2[r];
        drow[(size_t)m * LDW + 48 + l16] = c3[r];
    }
}

// ---------------- edge gather + weighted sum + atomic scatter ----------------
// One wave per edge; lane handles output pair {2*lane, 2*lane+1} of H=64.
__global__ __launch_bounds__(256)
void edge_msg(const float* __restrict__ xw, const float* __restrict__ basis,
              const int* __restrict__ widx, const int* __restrict__ src,
              const int* __restrict__ dst, float* __restrict__ agg) {
    int e = blockIdx.x * 8 + (threadIdx.x >> 5);
    int lane = threadIdx.x & 31;
    int sn = src[e];
    int dn = dst[e];
    const float* base = xw + (size_t)sn * LDW;
    float ax = 0.0f, ay = 0.0f;
#pragma unroll
    for (int s = 0; s < 4; ++s) {
        float w = basis[e * 4 + s];
        const float* g = base + widx[e * 4 + s] * CH + lane * 2;
        ax += w * g[0];
        ay += w * g[1];
    }
    float* out = agg + (size_t)dn * CH + lane * 2;
    __hip_atomic_fetch_add(&out[0], ax, __ATOMIC_RELAXED, __HIP_MEMORY_SCOPE_AGENT);
    __hip_atomic_fetch_add(&out[1], ay, __ATOMIC_RELAXED, __HIP_MEMORY_SCOPE_AGENT);
}

// conv_out = agg * inv_deg + xw_root_block + bias
__global__ __launch_bounds__(256)
void finalize_conv(const float* __restrict__ agg, const float* __restrict__ invdeg,
                   const float* __restrict__ xw, const float* __restrict__ bias,
                   float* __restrict__ out) {
    int i = blockIdx.x * 256 + threadIdx.x;  // < N*H
    int n = i >> 6;
    int o = i & 63;
    out[i] = agg[i] * invdeg[n] + xw[(size_t)n * LDW + K25 * CH + o] + bias[o];
}

// fused GRU cell update + output write
__global__ __launch_bounds__(256)
void gru_update(const float* __restrict__ xcr, const float* __restrict__ xcz,
                const float* __restrict__ xcn, const float* __restrict__ hrb,
                const float* __restrict__ hzb, float* __restrict__ h,
                float* __restrict__ out, int t) {
    int i = blockIdx.x * 256 + threadIdx.x;  // < N*H, i = n*H + o
    float hr = hrb[i];
    float r  = 1.0f / (1.0f + __expf(-(xcr[i] + hr)));
    float z  = 1.0f / (1.0f + __expf(-(xcz[i] + hzb[i])));
    float nn = tanhf(xcn[i] + r * hr);
    float hnew = (1.0f - z) * nn + z * h[i];
    h[i] = hnew;
    out[(size_t)i * CT + t] = hnew;                       // layer_output [B,G,H,T]
    if (t == CT - 1) out[(size_t)CN * CH * CT + i] = hnew; // last_state [1,B,G,H]
}

// ---------------- host orchestration ----------------
extern "C" void kernel_launch(void* const* d_in, const int* in_sizes, int n_in,
                              void* d_out, int out_size, void* d_ws, size_t ws_size,
                              hipStream_t stream) {
    const float* x      = (const float*)d_in[0];
    const int*   eidx   = (const int*)d_in[1];
    const float* eattr  = (const float*)d_in[2];
    const float* W_xr   = (const float*)d_in[3];
    const float* R_xr   = (const float*)d_in[4];
    const float* b_xr   = (const float*)d_in[5];
    const float* W_hr   = (const float*)d_in[6];
    const float* R_hr   = (const float*)d_in[7];
    const float* b_hr   = (const float*)d_in[8];
    const float* W_xz   = (const float*)d_in[9];
    const float* R_xz   = (const float*)d_in[10];
    const float* b_xz   = (const float*)d_in[11];
    const float* W_hz   = (const float*)d_in[12];
    const float* R_hz   = (const float*)d_in[13];
    const float* b_hz   = (const float*)d_in[14];
    const float* W_xn   = (const float*)d_in[15];
    const float* R_xn   = (const float*)d_in[16];
    const float* b_xn   = (const float*)d_in[17];
    float* out = (float*)d_out;

    // carve workspace
    char* p = (char*)d_ws;
    auto alloc = [&](size_t bytes) -> void* {
        void* r = (void*)p;
        p += (bytes + 255) & ~(size_t)255;
        return r;
    };
    int*   src   = (int*)  alloc((size_t)CE * 4);
    int*   dst   = (int*)  alloc((size_t)CE * 4);
    int*   widx  = (int*)  alloc((size_t)CE * 4 * 4);
    float* basis = (float*)alloc((size_t)CE * 4 * 4);
    float* deg   = (float*)alloc((size_t)CN * 4);
    float* xs    = (float*)alloc((size_t)CT * CN * CFIN * 4);
    float* wcxr  = (float*)alloc((size_t)CFIN * LDW * 4);
    float* wcxz  = (float*)alloc((size_t)CFIN * LDW * 4);
    float* wcxn  = (float*)alloc((size_t)CFIN * LDW * 4);
    float* wchr  = (float*)alloc((size_t)CH * LDW * 4);
    float* wchz  = (float*)alloc((size_t)CH * LDW * 4);
    float* xw    = (float*)alloc((size_t)CN * LDW * 4);     // ~109 MB, L2-resident
    float* agg   = (float*)alloc((size_t)CN * CH * 4);
    float* xcr   = (float*)alloc((size_t)CT * CN * CH * 4);
    float* xcz   = (float*)alloc((size_t)CT * CN * CH * 4);
    float* xcn   = (float*)alloc((size_t)CT * CN * CH * 4);
    float* hrb   = (float*)alloc((size_t)CN * CH * 4);
    float* hzb   = (float*)alloc((size_t)CN * CH * 4);
    float* h     = (float*)alloc((size_t)CN * CH * 4);

    const int NH = CN * CH;                 // 1,048,576
    const dim3 gemmGrid(CN / 128, LDW / 64); // (128, 26)

    // ---- prep ----
    zero_f32<<<(CN + 255) / 256, 256, 0, stream>>>(deg, CN);
    zero_f32<<<(NH + 255) / 256, 256, 0, stream>>>(h, NH);
    prep_edges<<<CE / 256, 256, 0, stream>>>(eidx, eattr, src, dst, basis, widx, deg);
    inv_deg_k<<<(CN + 255) / 256, 256, 0, stream>>>(deg);
    transpose_x<<<(CT * CN * CFIN) / 256, 256, 0, stream>>>(x, xs);
    repack_w<<<(CFIN * LDW) / 256, 256, 0, stream>>>(W_xr, R_xr, wcxr, CFIN);
    repack_w<<<(CFIN * LDW) / 256, 256, 0, stream>>>(W_xz, R_xz, wcxz, CFIN);
    repack_w<<<(CFIN * LDW) / 256, 256, 0, stream>>>(W_xn, R_xn, wcxn, CFIN);
    repack_w<<<(CH * LDW) / 256, 256, 0, stream>>>(W_hr, R_hr, wchr, CH);
    repack_w<<<(CH * LDW) / 256, 256, 0, stream>>>(W_hz, R_hz, wchz, CH);

    // ---- x-side convolutions for all t (parallel phase) ----
    const float* wcs[3]  = {wcxr, wcxz, wcxn};
    const float* bxs[3]  = {b_xr, b_xz, b_xn};
    float*       xcs[3]  = {xcr, xcz, xcn};
    for (int t = 0; t < CT; ++t) {
        const float* xt = xs + (size_t)t * CN * CFIN;
        for (int c = 0; c < 3; ++c) {
            gemm_wmma<CFIN><<<gemmGrid, 256, 0, stream>>>(xt, wcs[c], xw);
            zero_f32<<<NH / 256, 256, 0, stream>>>(agg, NH);
            edge_msg<<<CE / 8, 256, 0, stream>>>(xw, basis, widx, src, dst, agg);
            finalize_conv<<<NH / 256, 256, 0, stream>>>(agg, deg, xw, bxs[c],
                                                        xcs[c] + (size_t)t * NH);
        }
    }

    // ---- sequential GRU recurrence ----
    for (int t = 0; t < CT; ++t) {
        gemm_wmma<CH><<<gemmGrid, 256, 0, stream>>>(h, wchr, xw);
        zero_f32<<<NH / 256, 256, 0, stream>>>(agg, NH);
        edge_msg<<<CE / 8, 256, 0, stream>>>(xw, basis, widx, src, dst, agg);
        finalize_conv<<<NH / 256, 256, 0, stream>>>(agg, deg, xw, b_hr, hrb);

        gemm_wmma<CH><<<gemmGrid, 256, 0, stream>>>(h, wchz, xw);
        zero_f32<<<NH / 256, 256, 0, stream>>>(agg, NH);
        edge_msg<<<CE / 8, 256, 0, stream>>>(xw, basis, widx, src, dst, agg);
        finalize_conv<<<NH / 256, 256, 0, stream>>>(agg, deg, xw, b_hz, hzb);

        gru_update<<<NH / 256, 256, 0, stream>>>(xcr + (size_t)t * NH,
                                                 xcz + (size_t)t * NH,
                                                 xcn + (size_t)t * NH,
                                                 hrb, hzb, h, out, t);
    }
}